// TtMambaSSM_69080253989434
// MI455X (gfx1250) — compile-verified
//
#include <hip/hip_runtime.h>

typedef float v2f __attribute__((ext_vector_type(2)));
typedef float v8f __attribute__((ext_vector_type(8)));

#define D_INNER 5120
#define DT_RANK 160
#define BATCH   128
#define NSTATE  16
#define NPAD    32

// workspace layout (in floats)
#define WS_XR   0                          // 128*160
#define WS_BP   (WS_XR + BATCH*DT_RANK)    // 128*16
#define WS_CP   (WS_BP + BATCH*NSTATE)     // 128*16
#define WS_BC   (WS_CP + BATCH*NSTATE)     // 128
#define WS_ANEG (WS_BC + BATCH)            // 5120*16

// ---------------------------------------------------------------------------
// Kernel 1: x-projections via fp32 WMMA.
//   nt 0..9  -> xr tile (delta_t_proj_w, ncols=160)
//   nt 10    -> Bp      (B_proj_w, ncols=16)
//   nt 11    -> Cp      (C_proj_w, ncols=16)
// 4 waves split K=5120 into 1280-stripes; LDS reduction.
// ---------------------------------------------------------------------------
__global__ __launch_bounds__(128) void proj_kernel(
    const float* __restrict__ x,
    const float* __restrict__ dtr_w,
    const float* __restrict__ b_w,
    const float* __restrict__ c_w,
    float* __restrict__ ws)
{
    __shared__ float red[4 * 256];
    const int wave = threadIdx.x >> 5;
    const int lane = threadIdx.x & 31;
    const int half = lane >> 4;
    const int l16  = lane & 15;
    const int nt   = blockIdx.x;   // 0..11
    const int bt   = blockIdx.y;   // 0..7

    const float* W; int ncols; int n0; float* out;
    if (nt < 10)       { W = dtr_w; ncols = DT_RANK; n0 = nt * 16; out = ws + WS_XR; }
    else if (nt == 10) { W = b_w;   ncols = NSTATE;  n0 = 0;       out = ws + WS_BP; }
    else               { W = c_w;   ncols = NSTATE;  n0 = 0;       out = ws + WS_CP; }

    const int b0 = bt * 16;
    const float* xrow = x + (size_t)(b0 + l16) * D_INNER;  // A-frag row for this lane

    v8f acc = {};
    const int kbeg = wave * (D_INNER / 4);
    const int kend = kbeg + (D_INNER / 4);
    for (int k0 = kbeg; k0 < kend; k0 += 4) {
        const int ka = k0 + 2 * half;
        v2f a, b;
        a.x = xrow[ka];
        a.y = xrow[ka + 1];
        b.x = W[(size_t)ka       * ncols + n0 + l16];
        b.y = W[(size_t)(ka + 1) * ncols + n0 + l16];
        acc = __builtin_amdgcn_wmma_f32_16x16x4_f32(false, a, false, b,
                                                    (short)0, acc, false, false);
    }

    #pragma unroll
    for (int r = 0; r < 8; ++r) {
        const int m = r + 8 * half;
        red[wave * 256 + m * 16 + l16] = acc[r];
    }
    __syncthreads();
    if (wave == 0) {
        #pragma unroll
        for (int r = 0; r < 8; ++r) {
            const int m = r + 8 * half;
            const int idx = m * 16 + l16;
            const float s = red[idx] + red[256 + idx] + red[512 + idx] + red[768 + idx];
            out[(size_t)(b0 + m) * ncols + n0 + l16] = s;
        }
    }
}

// ---------------------------------------------------------------------------
// Kernel 2 (prep): A = -exp(A_log); BC[b] = dot(Bp[b], Cp[b])
// ---------------------------------------------------------------------------
__global__ void prep_kernel(const float* __restrict__ A_log, float* __restrict__ ws)
{
    const int tid = blockIdx.x * blockDim.x + threadIdx.x;
    const int na = D_INNER * NSTATE;   // 81920
    if (tid < na) {
        ws[WS_ANEG + tid] = -__expf(A_log[tid]);
    } else if (tid < na + BATCH) {
        const int b = tid - na;
        const float* Bp = ws + WS_BP + b * NSTATE;
        const float* Cp = ws + WS_CP + b * NSTATE;
        float s = 0.f;
        #pragma unroll
        for (int k = 0; k < NSTATE; ++k) s += Bp[k] * Cp[k];
        ws[WS_BC + b] = s;
    }
}

// ---------------------------------------------------------------------------
// Kernel 3: fused dt-GEMM (K=160, fp32 WMMA) + softplus + SSM step + C-reduce.
// One wave owns a 16(batch) x 16(d_inner) tile; lane's 8 accumulator elements
// map to (b,d) pairs; each loads its 64B h row (4 x float4, 128B aligned).
// ---------------------------------------------------------------------------
__global__ __launch_bounds__(128) void ssm_kernel(
    const float* __restrict__ x,
    const float* __restrict__ h,
    const float* __restrict__ dt_w,    // (160, 5120)
    const float* __restrict__ dt_b,    // (5120)
    const float* __restrict__ Dw,      // (5120)
    const float* __restrict__ ws,
    float* __restrict__ y)
{
    const int wave = threadIdx.x >> 5;
    const int lane = threadIdx.x & 31;
    const int half = lane >> 4;
    const int l16  = lane & 15;
    const int nt   = blockIdx.x * 4 + wave;   // 0..319
    const int bt   = blockIdx.y;              // 0..7
    const int d0   = nt * 16;
    const int b0   = bt * 16;
    const int d    = d0 + l16;

    const float* xrrow = ws + WS_XR + (size_t)(b0 + l16) * DT_RANK;

    v8f acc = {};
    #pragma unroll 4
    for (int k0 = 0; k0 < DT_RANK; k0 += 4) {
        const int ka = k0 + 2 * half;
        v2f a, b;
        a.x = xrrow[ka];
        a.y = xrrow[ka + 1];
        b.x = dt_w[(size_t)ka       * D_INNER + d];
        b.y = dt_w[(size_t)(ka + 1) * D_INNER + d];
        acc = __builtin_amdgcn_wmma_f32_16x16x4_f32(false, a, false, b,
                                                    (short)0, acc, false, false);
    }

    // lane-invariant per-d values (same d for all 8 elements of this lane)
    const float bias = dt_b[d];
    const float Dd   = Dw[d];
    const float4* ap = (const float4*)(ws + WS_ANEG + (size_t)d * NSTATE);
    float4 An[4];
    An[0] = ap[0]; An[1] = ap[1]; An[2] = ap[2]; An[3] = ap[3];

    #pragma unroll
    for (int r = 0; r < 8; ++r) {
        const int b = b0 + r + 8 * half;
        const float z  = acc[r] + bias;
        const float zc = fminf(z, 20.f);
        const float dtv = (z > 20.f) ? z : __logf(1.f + __expf(zc));

        const float xv = x[(size_t)b * D_INNER + d];
        const float4* hp = (const float4*)(h + ((size_t)b * D_INNER + d) * NPAD);
        const float4* cp = (const float4*)(ws + WS_CP + b * NSTATE);

        float sum = 0.f;
        #pragma unroll
        for (int q = 0; q < 4; ++q) {
            const float4 hv = hp[q];
            const float4 cv = cp[q];
            const float4 av = An[q];
            sum += __expf(dtv * av.x) * hv.x * cv.x;
            sum += __expf(dtv * av.y) * hv.y * cv.y;
            sum += __expf(dtv * av.z) * hv.z * cv.z;
            sum += __expf(dtv * av.w) * hv.w * cv.w;
        }
        const float bc = ws[WS_BC + b];
        y[(size_t)b * D_INNER + d] = sum + xv * (dtv * bc + Dd);
    }
}

extern "C" void kernel_launch(void* const* d_in, const int* in_sizes, int n_in,
                              void* d_out, int out_size, void* d_ws, size_t ws_size,
                              hipStream_t stream) {
    const float* x     = (const float*)d_in[0];
    const float* h     = (const float*)d_in[1];
    const float* dtr_w = (const float*)d_in[2];
    const float* dt_w  = (const float*)d_in[3];
    const float* dt_b  = (const float*)d_in[4];
    const float* b_w   = (const float*)d_in[5];
    const float* c_w   = (const float*)d_in[6];
    const float* A_log = (const float*)d_in[7];
    const float* Dw    = (const float*)d_in[8];
    float* ws = (float*)d_ws;
    float* y  = (float*)d_out;

    proj_kernel<<<dim3(12, 8), 128, 0, stream>>>(x, dtr_w, b_w, c_w, ws);
    prep_kernel<<<(D_INNER * NSTATE + BATCH + 255) / 256, 256, 0, stream>>>(A_log, ws);
    ssm_kernel<<<dim3(80, 8), 128, 0, stream>>>(x, h, dt_w, dt_b, Dw, ws, y);
}